// DiffLogicLayer_89824946029085
// MI455X (gfx1250) — compile-verified
//
#include <hip/hip_runtime.h>

#define IN_DIM   8192
#define OUT_DIM  8192
#define NTHREADS 1024
#define RPB      4          // batch rows staged per block (4 x 32KB = 128KB LDS)

typedef __attribute__((ext_vector_type(2))) float v2f;
typedef __attribute__((ext_vector_type(8))) float v8f;
typedef __attribute__((ext_vector_type(4))) int   v4i;

// Each of the 16 relaxed gates expressed in the basis {1, a, b, a*b}.
__constant__ float MTAB[16][4] = {
    {0.f, 0.f, 0.f, 0.f},   //  0: False
    {0.f, 0.f, 0.f, 1.f},   //  1: a AND b
    {0.f, 1.f, 0.f,-1.f},   //  2: a AND NOT b
    {0.f, 1.f, 0.f, 0.f},   //  3: a
    {0.f, 0.f, 1.f,-1.f},   //  4: NOT a AND b
    {0.f, 0.f, 1.f, 0.f},   //  5: b
    {0.f, 1.f, 1.f,-2.f},   //  6: XOR
    {0.f, 1.f, 1.f,-1.f},   //  7: OR
    {1.f,-1.f,-1.f, 1.f},   //  8: NOR
    {1.f,-1.f,-1.f, 2.f},   //  9: XNOR
    {1.f, 0.f,-1.f, 0.f},   // 10: NOT b
    {1.f, 0.f,-1.f, 1.f},   // 11: a OR NOT b
    {1.f,-1.f, 0.f, 0.f},   // 12: NOT a
    {1.f,-1.f, 0.f, 1.f},   // 13: NOT a OR b
    {1.f, 0.f, 0.f,-1.f},   // 14: NAND
    {1.f, 0.f, 0.f, 0.f},   // 15: True
};

#if defined(__HIP_DEVICE_COMPILE__) && \
    __has_builtin(__builtin_amdgcn_global_load_async_to_lds_b128) && \
    __has_builtin(__builtin_amdgcn_s_wait_asynccnt)
#define USE_ASYNC_LDS 1
#endif

#ifdef USE_ASYNC_LDS
// Builtin takes pointers to 16-byte int vectors (per hipcc diagnostic).
typedef __attribute__((address_space(1))) v4i g_v4i;
typedef __attribute__((address_space(3))) v4i l_v4i;
// AS3 pointers are 32-bit LDS offsets; flat->LDS keeps addr[31:0] (ISA 10.2).
__device__ static inline g_v4i* to_gbl(const void* p) {
    return (g_v4i*)(unsigned long long)p;
}
__device__ static inline l_v4i* to_lds(const void* p) {
    return (l_v4i*)(unsigned int)(unsigned long long)p;
}
#endif

// ---------------------------------------------------------------------------
// Kernel 1: c[o][0..3] = softmax(w[o]) . MTAB  via V_WMMA_F32_16X16X4_F32.
// One wave per 16 output neurons. Lane-variant operands are re-materialized
// with runtime-addressed *loads* (not register-array indexing) to avoid
// v_cndmask select chains.
// ISA layouts (CDNA5 ISA): A 16x4: lanes 0-15 M=0..15, v0=K(2h+0), v1=K(2h+1)
// with h = lane>>4 selecting K-pair. B 4x16: v0=row K(2h), v1=row K(2h+1),
// N = lane&15. D: vgpr r holds M = r + 8h, N = lane&15.
// ---------------------------------------------------------------------------
__global__ __launch_bounds__(32) void difflogic_coef_kernel(
    const float* __restrict__ w, float* __restrict__ coef) {
    const int lane  = threadIdx.x;        // 0..31, wave32
    const int rowm  = lane & 15;          // A-matrix row (M) this lane feeds
    const int h     = lane >> 4;          // K-half selector
    const int obase = blockIdx.x * 16;

    const float* wr = w + (size_t)(obase + rowm) * 16;
    float mx = wr[0];
#pragma unroll
    for (int k = 1; k < 16; ++k) mx = fmaxf(mx, wr[k]);
    float s = 0.f;
#pragma unroll
    for (int k = 0; k < 16; ++k) s += __expf(wr[k] - mx);
    const float inv = 1.f / s;

#if defined(__HIP_DEVICE_COMPILE__) && __has_builtin(__builtin_amdgcn_wmma_f32_16x16x4_f32)
    const int   n     = rowm;             // B/D column (N) this lane holds
    const int   nc    = n & 3;
    const float bmask = (n < 4) ? 1.f : 0.f;
    const float* wh   = wr + 2 * h;                 // runtime base: K-pair
    const float* mh   = &MTAB[0][0] + (2 * h) * 4;  // runtime base: MTAB rows

    v8f d = {0.f, 0.f, 0.f, 0.f, 0.f, 0.f, 0.f, 0.f};
#pragma unroll
    for (int t = 0; t < 4; ++t) {         // K = 16 as 4 chunks of 4
        v2f A;
        A.x = __expf(wh[4 * t + 0] - mx) * inv;     // e[4t+2h+0]/sum
        A.y = __expf(wh[4 * t + 1] - mx) * inv;     // e[4t+2h+1]/sum
        v2f B;
        B.x = mh[(4 * t + 0) * 4 + nc] * bmask;     // MTAB[4t+2h+0][n]
        B.y = mh[(4 * t + 1) * 4 + nc] * bmask;     // MTAB[4t+2h+1][n]
        d = __builtin_amdgcn_wmma_f32_16x16x4_f32(
                false, A, false, B, (short)0, d, false, false);
    }
    if (n < 4) {
#pragma unroll
        for (int r = 0; r < 8; ++r)
            coef[(size_t)(obase + r + 8 * h) * 4 + n] = d[r];
    }
#else
    // Scalar fallback (also taken by the host pass).
    if (h == 0) {
        float c0 = 0.f, c1 = 0.f, c2 = 0.f, c3 = 0.f;
#pragma unroll
        for (int k = 0; k < 16; ++k) {
            const float p = __expf(wr[k] - mx) * inv;
            c0 += p * MTAB[k][0]; c1 += p * MTAB[k][1];
            c2 += p * MTAB[k][2]; c3 += p * MTAB[k][3];
        }
        float* cp = coef + (size_t)(obase + rowm) * 4;
        cp[0] = c0; cp[1] = c1; cp[2] = c2; cp[3] = c3;
    }
#endif
}

// ---------------------------------------------------------------------------
// Kernel 2: each block stages RPB consecutive batch rows (contiguous 128KB of
// x) into LDS with async b128 copies, then streams the 8192 neurons once:
// idx/coef are loaded once per RPB rows (4x less L2 traffic), gathers hit the
// 64-bank LDS, results go out via non-temporal stores.
// ---------------------------------------------------------------------------
__global__ __launch_bounds__(NTHREADS) void difflogic_gate_kernel(
    const float* __restrict__ x,
    const int*   __restrict__ ia,
    const int*   __restrict__ ib,
    const float4* __restrict__ coef,
    float* __restrict__ out) {
    __shared__ float4 stile[RPB * IN_DIM / 4];   // 128 KB
    const int    tid   = threadIdx.x;
    const size_t rbase = (size_t)blockIdx.x * RPB;
    const float4* xr   = (const float4*)(x + rbase * IN_DIM);

#pragma unroll
    for (int kk = 0; kk < (RPB * IN_DIM / 4) / NTHREADS; ++kk) {  // 8 iters
        const int i = tid + kk * NTHREADS;
#ifdef USE_ASYNC_LDS
        __builtin_amdgcn_global_load_async_to_lds_b128(
            to_gbl(xr + i), to_lds(stile + i), 0, 0);
#else
        stile[i] = xr[i];
#endif
    }
#ifdef USE_ASYNC_LDS
    __builtin_amdgcn_s_wait_asynccnt(0);
#endif
    __syncthreads();

    const float* S = (const float*)stile;
#pragma unroll
    for (int kk = 0; kk < OUT_DIM / NTHREADS; ++kk) {             // 8 iters
        const int    o   = tid + kk * NTHREADS;
        const int    a_i = ia[o];
        const int    b_i = ib[o];
        const float4 c   = coef[o];
#pragma unroll
        for (int r = 0; r < RPB; ++r) {
            const float a = S[r * IN_DIM + a_i];
            const float b = S[r * IN_DIM + b_i];
            const float v = fmaf(c.w, a * b, fmaf(c.z, b, fmaf(c.y, a, c.x)));
            __builtin_nontemporal_store(v, out + (rbase + r) * OUT_DIM + o);
        }
    }
}

extern "C" void kernel_launch(void* const* d_in, const int* in_sizes, int n_in,
                              void* d_out, int out_size, void* d_ws, size_t ws_size,
                              hipStream_t stream) {
    (void)n_in; (void)out_size; (void)ws_size;
    const float* x  = (const float*)d_in[0];   // (BATCH, IN_DIM) f32
    const float* w  = (const float*)d_in[1];   // (OUT_DIM, 16)   f32
    const int*   ia = (const int*)d_in[2];     // (OUT_DIM,)      i32
    const int*   ib = (const int*)d_in[3];     // (OUT_DIM,)      i32
    float* out  = (float*)d_out;
    float* coef = (float*)d_ws;                // OUT_DIM * 4 floats = 128 KB

    const int batch = in_sizes[0] / IN_DIM;    // 2048

    difflogic_coef_kernel<<<OUT_DIM / 16, 32, 0, stream>>>(w, coef);
    difflogic_gate_kernel<<<batch / RPB, NTHREADS, 0, stream>>>(
        x, ia, ib, (const float4*)coef, out);
}